// DeepPolyConv2D_40183714021524
// MI455X (gfx1250) — compile-verified
//
#include <hip/hip_runtime.h>
#include <cstdint>

// ---- fixed module configuration (from reference) ----
static constexpr int O_CH = 16, C_IN = 3, KH = 4, KW = 4;
static constexpr int H_IMG = 48, W_IMG = 48;
static constexpr int HO = 24, WO = 24;                       // (48 + 2*1 - 4)/2 + 1
static constexpr int N_COLS = C_IN * H_IMG * W_IMG;          // 6912
static constexpr int N_ROWS = O_CH * HO * WO;                // 9216
static constexpr long long W_ELEMS = (long long)N_ROWS * N_COLS;  // 63,700,992
static constexpr long long W_VEC4  = W_ELEMS / 4;                  // 15,925,248

typedef float float4v __attribute__((ext_vector_type(4)));
typedef int   int4v   __attribute__((ext_vector_type(4)));

#if __has_builtin(__builtin_amdgcn_global_store_async_from_lds_b128)
#define HAVE_ASYNC_STORE 1
typedef __attribute__((address_space(1))) int4v g_int4;  // global int4
typedef __attribute__((address_space(3))) int4v l_int4;  // LDS int4
#else
#define HAVE_ASYNC_STORE 0
#endif

// -------------------------------------------------------------------------
// Pass 1: zero-fill the 255 MB W_mat region with 128-bit stores.
// Primary path: CDNA5 async LDS->global stores (ASYNCcnt, DMA-style, frees
// VALU, up to 63 outstanding per wave). Fallback: non-temporal b128 stores
// (output > L2 capacity and never re-read on device, so NT is the right TH).
// -------------------------------------------------------------------------
__global__ void fill_w_zero(float4v* __restrict__ out, long long nvec) {
  long long i      = (long long)blockIdx.x * blockDim.x + threadIdx.x;
  long long stride = (long long)gridDim.x * blockDim.x;
  float4v z = {0.f, 0.f, 0.f, 0.f};
#if HAVE_ASYNC_STORE
  __shared__ float4v zbuf[256];           // one 16B zero tile per thread
  zbuf[threadIdx.x] = z;
  __syncthreads();
  // AS(3) pointer: low 32 bits of the generic shared address are the LDS offset.
  l_int4* lsrc = (l_int4*)(uint32_t)(uintptr_t)&zbuf[threadIdx.x];
  for (; i < nvec; i += stride) {
    __builtin_amdgcn_global_store_async_from_lds_b128(
        (g_int4*)(uintptr_t)(out + i), lsrc, /*imm offset*/0, /*cpol*/0);
  }
#if __has_builtin(__builtin_amdgcn_s_wait_asynccnt)
  __builtin_amdgcn_s_wait_asynccnt(0);
#else
  asm volatile("s_wait_asynccnt 0" ::: "memory");
#endif
#else
  for (; i < nvec; i += stride) {
    __builtin_nontemporal_store(z, out + i);   // global_store_b128 th:NT
  }
#endif
}

// -------------------------------------------------------------------------
// Pass 2: scatter the conv weights into W_mat and write the bias tail.
// One thread per (row, c, kh, kw): 9216 * 48 = 442,368 threads, ~1.7 MB of
// scattered 4B stores. Thread e==0 of each row also writes bias[o].
// Stream ordering guarantees this runs after the zero-fill completes.
// -------------------------------------------------------------------------
__global__ void scatter_weights(const float* __restrict__ kern,
                                const float* __restrict__ bias,
                                const int* __restrict__ stride_p,
                                const int* __restrict__ pad_p,
                                float* __restrict__ out) {
  int tid = blockIdx.x * blockDim.x + threadIdx.x;
  constexpr int EPR = C_IN * KH * KW;                 // 48 entries per row
  if (tid >= N_ROWS * EPR) return;

  int r  = tid / EPR;                                 // output row (o,ho,wo)
  int e  = tid % EPR;
  int c  = e / (KH * KW);
  int kh = (e / KW) % KH;
  int kw = e % KW;

  int o   = r / (HO * WO);
  int rem = r % (HO * WO);
  int ho  = rem / WO;
  int wo  = rem % WO;

  int s = stride_p[0];
  int p = pad_p[0];

  if (e == 0) out[W_ELEMS + r] = bias[o];             // b_vec tail

  int h = ho * s - p + kh;
  int w = wo * s - p + kw;
  if ((unsigned)h < (unsigned)H_IMG && (unsigned)w < (unsigned)W_IMG) {
    long long col = ((long long)c * H_IMG + h) * W_IMG + w;
    out[(long long)r * N_COLS + col] =
        kern[((o * C_IN + c) * KH + kh) * KW + kw];
  }
}

extern "C" void kernel_launch(void* const* d_in, const int* in_sizes, int n_in,
                              void* d_out, int out_size, void* d_ws, size_t ws_size,
                              hipStream_t stream) {
  const float* kern     = (const float*)d_in[0];  // [16,3,4,4]
  const float* bias     = (const float*)d_in[1];  // [16]
  const int*   stride_p = (const int*)d_in[2];    // scalar 2
  const int*   pad_p    = (const int*)d_in[3];    // scalar 1
  float* out = (float*)d_out;                     // [9216*6912 + 9216] f32

  // Pass 1: ~2M threads, ~7.6 b128 stores each -> enough waves to saturate HBM.
  fill_w_zero<<<8192, 256, 0, stream>>>((float4v*)out, W_VEC4);

  // Pass 2: 442,368 threads = 1728 blocks of 256.
  scatter_weights<<<(N_ROWS * C_IN * KH * KW + 255) / 256, 256, 0, stream>>>(
      kern, bias, stride_p, pad_p, out);
}